// ProductResidualVectorQuantize_12137577578695
// MI455X (gfx1250) — compile-verified
//
#include <hip/hip_runtime.h>
#include <hip/hip_bf16.h>

// ---------------- constants (match reference) ----------------
#define Hh   6
#define Cc   64
#define OV   4
#define Gg   3
#define NVQ  6
#define Kk   1024
#define Dd   8
#define Bb   32
#define Ww   2400
#define Tt   600           // Ww / OV
#define GD   512
#define WC   (Ww*Cc)       // 153600
#define HWC  (Hh*Ww*Cc)    // 921600
#define ZQN  (Bb*Hh*Ww*Cc) // 29491200
#define IDXN (Bb*NVQ*Gg*Tt)// 345600
#define NTOK (Bb*Tt)       // 19200 tokens per group
#define TILES_PER_BLOCK 8  // 8 waves * 16 tokens = 128 tokens / block
#define BLOCKS_X (NTOK / (TILES_PER_BLOCK*16))  // 150
#define NPART (BLOCKS_X * Gg)                   // 450
#define LOSS_SCALE (1.0f / (float)(NTOK * Dd * Gg)) // 1/460800

typedef float v2f __attribute__((ext_vector_type(2)));
typedef float v8f __attribute__((ext_vector_type(8)));

// IEEE float -> totally-ordered unsigned (monotone: a<b  <=>  map(a)<map(b))
__device__ __forceinline__ unsigned int sortable_f32(float f) {
  unsigned int u = __float_as_uint(f);
  return u ^ (((unsigned int)((int)u >> 31)) | 0x80000000u);
}

// ---------------- kernel 1: normalize codebooks -> en in workspace ----------
__global__ void prvq_normalize_cb(const float* __restrict__ cbs,
                                  float* __restrict__ en) {
  int v = blockIdx.x * 256 + threadIdx.x;           // vector id, G*NVQ*K total
  if (v >= Gg * NVQ * Kk) return;
  const float4* src = (const float4*)(cbs + (size_t)v * Dd);
  float4 a = src[0], b = src[1];
  float n2 = a.x*a.x + a.y*a.y + a.z*a.z + a.w*a.w +
             b.x*b.x + b.y*b.y + b.z*b.z + b.w*b.w;
  float s = 1.0f / fmaxf(sqrtf(n2), 1e-12f);
  float4* dst = (float4*)(en + (size_t)v * Dd);
  a.x *= s; a.y *= s; a.z *= s; a.w *= s;
  b.x *= s; b.y *= s; b.z *= s; b.w *= s;
  dst[0] = a; dst[1] = b;
}

// ---------------- kernel 2: main PRVQ ----------------
__global__ __launch_bounds__(256)
void prvq_main(const float* __restrict__ zin,      // (B, H*W, C)
               const float* __restrict__ pdw,      // (G, D, GD)
               const float* __restrict__ puw,      // (G, GD, D)
               const float* __restrict__ en,       // (G, NVQ, K, D) normalized
               float* __restrict__ outF,           // zq | idx | cm | cb
               float* __restrict__ partials) {
  __shared__ __align__(16) float pdS[GD * Dd];     // transposed: [j][d]
  __shared__ __align__(16) float puS[GD * Dd];     // [j][d]
  __shared__ __align__(16) float residS[8][16][Dd];
  __shared__ __align__(16) float zqS[8][16][Dd];
  __shared__ int   codesS[8][16];
  __shared__ float wsum[8];

  const int tid  = threadIdx.x;
  const int wv   = tid >> 5;
  const int lane = tid & 31;
  const int half = lane >> 4;      // 0: lanes 0-15, 1: lanes 16-31
  const int col  = lane & 15;      // N column / M row slot
  const int g    = blockIdx.y;
  const int tile_base = (blockIdx.x * TILES_PER_BLOCK + wv) * 16;

  // ---- stage pd (transposed) and pu into LDS ----
  {
    const float* pdg = pdw + (size_t)g * Dd * GD;  // [d][j]
    const float* pug = puw + (size_t)g * GD * Dd;  // [j][d]
    for (int e = tid; e < GD * Dd; e += 256) {
      int d = e >> 9;           // e / 512
      int j = e & 511;
      pdS[j * Dd + d] = pdg[e];
      puS[e] = pug[e];
    }
    // zero zq accumulators for this wave's tile
    for (int q = lane; q < 16 * Dd; q += 32) (&zqS[wv][0][0])[q] = 0.0f;
  }

  // ---- per-lane precompute of scatter/gather offsets (depends on g, lane) --
  int pre[16];
#pragma unroll
  for (int jj = 0; jj < 16; ++jj) {
    int j   = lane + 32 * jj;
    int idx = g * GD + j;        // global folded-feature index, < 2048
    int o   = idx / 384;
    int rem = idx - o * 384;
    int c   = rem / 6;
    int h   = rem - c * 6;
    pre[jj] = h * WC + o * Cc + c;
  }
  __syncthreads();

  // ---- step 1: proj_down -> residS[wv][m][0..7] ----
  for (int m = 0; m < 16; ++m) {
    int tok = tile_base + m;
    int b   = tok / Tt;
    int t   = tok - b * Tt;
    int base = b * HWC + t * (OV * Cc);
    float acc[Dd];
#pragma unroll
    for (int d = 0; d < Dd; ++d) acc[d] = 0.0f;
#pragma unroll
    for (int jj = 0; jj < 16; ++jj) {
      float xv = zin[base + pre[jj]];
      int j = lane + 32 * jj;
      float4 p0 = *(const float4*)&pdS[j * Dd];
      float4 p1 = *(const float4*)&pdS[j * Dd + 4];
      acc[0] += xv * p0.x; acc[1] += xv * p0.y;
      acc[2] += xv * p0.z; acc[3] += xv * p0.w;
      acc[4] += xv * p1.x; acc[5] += xv * p1.y;
      acc[6] += xv * p1.z; acc[7] += xv * p1.w;
    }
#pragma unroll
    for (int d = 0; d < Dd; ++d) {
      float a = acc[d];
      a += __shfl_xor(a, 16);
      a += __shfl_xor(a, 8);
      a += __shfl_xor(a, 4);
      a += __shfl_xor(a, 2);
      a += __shfl_xor(a, 1);
      if (lane == 0) residS[wv][m][d] = a;
    }
  }

  // ---- step 2: residual VQ, 6 streams, WMMA f32 16x16x4 for sim GEMM ----
  const float* en_g = en + (size_t)g * NVQ * Kk * Dd;
  float sq_acc = 0.0f;

  for (int i = 0; i < NVQ; ++i) {
    const float* en_gi = en_g + (size_t)i * Kk * Dd;
    __syncthreads();   // residS ready (proj_down or previous update)

    // A fragments: 16x8 residual tile, split into two K=4 WMMAs.
    // A layout (ISA 7.12.2): lanes 0-15 hold K=0,1 in v[0],v[1]; lanes 16-31 K=2,3.
    const float* rrow = &residS[wv][col][0];
    v2f a0 = *(const v2f*)(rrow + 2 * half);       // K = 0..3
    v2f a1 = *(const v2f*)(rrow + 4 + 2 * half);   // K = 4..7

    float bestv[8];
    int   bestc[8];
#pragma unroll
    for (int r = 0; r < 8; ++r) { bestv[r] = -__builtin_huge_valf(); bestc[r] = 0; }

    for (int n = 0; n < Kk / 16; ++n) {            // 64 column tiles of 16 codes
      int code = n * 16 + col;
      const float* brow = en_gi + (size_t)code * Dd;
      // B layout (4x16): lane = column, lanes 0-15 hold K=0,1; lanes 16-31 K=2,3.
      v2f b0 = *(const v2f*)(brow + 2 * half);
      v2f b1 = *(const v2f*)(brow + 4 + 2 * half);
      // speculative prefetch of the B tile 8 iterations ahead (clamped)
      {
        int pn = (n + 8 < Kk / 16) ? (n + 8) : (Kk / 16 - 1);
        __builtin_prefetch(en_gi + ((size_t)(pn * 16 + col)) * Dd, 0, 0);
      }
      v8f cacc = {0.f,0.f,0.f,0.f,0.f,0.f,0.f,0.f};
      cacc = __builtin_amdgcn_wmma_f32_16x16x4_f32(false, a0, false, b0,
                                                   (short)0, cacc, false, false);
      cacc = __builtin_amdgcn_wmma_f32_16x16x4_f32(false, a1, false, b1,
                                                   (short)0, cacc, false, false);
      // C layout: VGPR r -> row r (lanes 0-15) / row r+8 (lanes 16-31), N = col.
#pragma unroll
      for (int r = 0; r < 8; ++r) {
        float v = cacc[r];
        if (v > bestv[r]) { bestv[r] = v; bestc[r] = code; }  // strict >: lowest idx kept
      }
    }

    // Branchless argmax across the 16 columns of each half:
    // pack (sortable(sim) << 32) | ~code -> unsigned max picks larger sim,
    // and on ties the LOWER code (matches jnp.argmax first-max semantics).
#pragma unroll
    for (int r = 0; r < 8; ++r) {
      unsigned long long key =
          ((unsigned long long)sortable_f32(bestv[r]) << 32) |
          (unsigned long long)(unsigned int)(~bestc[r]);
#pragma unroll
      for (int off = 8; off >= 1; off >>= 1) {
        unsigned long long ok = __shfl_xor(key, off);
        key = (ok > key) ? ok : key;                 // branchless u64 max
      }
      if (col == 0) {            // lane 0 -> row r, lane 16 -> row r+8
        int bc = (int)(~(unsigned int)key);          // recover code
        int m = r + 8 * half;
        codesS[wv][m] = bc;
        int tok = tile_base + m;
        int b   = tok / Tt;
        int t   = tok - b * Tt;
        outF[ZQN + ((size_t)(b * NVQ + i) * Gg + g) * Tt + t] = (float)bc;
      }
    }
    __syncthreads();   // codesS visible to all lanes

    // update: residual -= q ; zq += q ; sq += |residual - q|^2
    if (lane < 16) {
      int m    = lane;
      int code = codesS[wv][m];
      const float4* q4 = (const float4*)(en_gi + (size_t)code * Dd);
      float4 q0 = q4[0], q1 = q4[1];
      float qv[Dd] = {q0.x, q0.y, q0.z, q0.w, q1.x, q1.y, q1.z, q1.w};
      float* rs = residS[wv][m];
      float* zs = zqS[wv][m];
#pragma unroll
      for (int d = 0; d < Dd; ++d) {
        float rv = rs[d] - qv[d];
        sq_acc += rv * rv;
        rs[d] = rv;
        zs[d] += qv[d];
      }
    }
  }
  __syncthreads();

  // ---- step 3: proj_up + scatter back to (B, H*W, C) ----
  for (int m = 0; m < 16; ++m) {
    int tok = tile_base + m;
    int b   = tok / Tt;
    int t   = tok - b * Tt;
    int base = b * HWC + t * (OV * Cc);
    float4 z0 = *(const float4*)&zqS[wv][m][0];
    float4 z1 = *(const float4*)&zqS[wv][m][4];
#pragma unroll
    for (int jj = 0; jj < 16; ++jj) {
      int j = lane + 32 * jj;
      float4 p0 = *(const float4*)&puS[j * Dd];
      float4 p1 = *(const float4*)&puS[j * Dd + 4];
      float up = z0.x*p0.x + z0.y*p0.y + z0.z*p0.z + z0.w*p0.w +
                 z1.x*p1.x + z1.y*p1.y + z1.z*p1.z + z1.w*p1.w;
      outF[base + pre[jj]] = up;
    }
  }

  // ---- loss partial (deterministic: per-block, no float atomics) ----
  float s = sq_acc;
  s += __shfl_xor(s, 16);
  s += __shfl_xor(s, 8);
  s += __shfl_xor(s, 4);
  s += __shfl_xor(s, 2);
  s += __shfl_xor(s, 1);
  if (lane == 0) wsum[wv] = s;
  __syncthreads();
  if (tid == 0) {
    float tot = 0.0f;
#pragma unroll
    for (int w = 0; w < 8; ++w) tot += wsum[w];
    partials[blockIdx.y * BLOCKS_X + blockIdx.x] = tot;
  }
}

// ---------------- kernel 3: finalize losses ----------------
__global__ void prvq_finalize(const float* __restrict__ partials,
                              float* __restrict__ outF) {
  __shared__ float sh[256];
  float s = 0.0f;
  for (int i = threadIdx.x; i < NPART; i += 256) s += partials[i];
  sh[threadIdx.x] = s;
  __syncthreads();
  for (int w = 128; w >= 1; w >>= 1) {
    if (threadIdx.x < w) sh[threadIdx.x] += sh[threadIdx.x + w];
    __syncthreads();
  }
  if (threadIdx.x == 0) {
    float v = sh[0] * LOSS_SCALE;
    outF[ZQN + IDXN]     = v;   // cm_loss
    outF[ZQN + IDXN + 1] = v;   // cb_loss (identical forward value)
  }
}

// ---------------- launch ----------------
extern "C" void kernel_launch(void* const* d_in, const int* in_sizes, int n_in,
                              void* d_out, int out_size, void* d_ws, size_t ws_size,
                              hipStream_t stream) {
  const float* z_e  = (const float*)d_in[0];
  const float* pdw  = (const float*)d_in[1];
  const float* puw  = (const float*)d_in[2];
  const float* cbs  = (const float*)d_in[3];
  // d_in[4] = num_streams (== 6, baked into NVQ)
  float* outF = (float*)d_out;
  float* ws   = (float*)d_ws;
  float* en       = ws;                         // G*NVQ*K*D = 147456 floats
  float* partials = ws + (size_t)Gg * NVQ * Kk * Dd;  // NPART floats

  int nvec = Gg * NVQ * Kk;                     // 18432 codebook vectors
  prvq_normalize_cb<<<(nvec + 255) / 256, 256, 0, stream>>>(cbs, en);

  dim3 grid(BLOCKS_X, Gg);
  prvq_main<<<grid, 256, 0, stream>>>(z_e, pdw, puw, en, outF, partials);

  prvq_finalize<<<1, 256, 0, stream>>>(partials, outF);
}